// CategoricalDnn_39324720562872
// MI455X (gfx1250) — compile-verified
//
#include <hip/hip_runtime.h>

#define B_SZ   65536
#define F_SZ   32
#define V_SZ   1001
#define D_SZ   64
#define N_NUMC 16
#define OUT_COLS (F_SZ * D_SZ + N_NUMC)   // 2064
#define BN_EPS 1e-5f

typedef float v8f __attribute__((ext_vector_type(8)));
typedef float v2f __attribute__((ext_vector_type(2)));
typedef float f4  __attribute__((ext_vector_type(4)));

// ---------------------------------------------------------------- kernel 1
__global__ void zero_counts_kernel(int* __restrict__ counts) {
    int i = blockIdx.x * blockDim.x + threadIdx.x;
    if (i < F_SZ * V_SZ) counts[i] = 0;
}

// ---------------------------------------------------------------- kernel 2
// Histogram of indices: count[f, v] = #{b : cat_idx[b, f] == v}.
// 2M atomics spread over 32K counters across L2 channels — cheap.
__global__ void hist_kernel(const int* __restrict__ cat_idx,
                            int* __restrict__ counts) {
    int i = blockIdx.x * blockDim.x + threadIdx.x;   // over B*F
    if (i < B_SZ * F_SZ) {
        int f = i & (F_SZ - 1);                      // row-major (B, F)
        int v = cat_idx[i];
        atomicAdd(&counts[f * V_SZ + v], 1);
    }
}

// ---------------------------------------------------------------- kernel 3
// Count-weighted stats via V_WMMA_F32_16X16X4_F32.
//   S1[f,d] = sum_v count[f,v] * t[f,v,d]
//   S2[f,d] = sum_v count[f,v] * t[f,v,d]^2
// One wave per (feature, 16-wide d tile). A row 0 = counts (other rows 0),
// B[k][n] = table values; D row 0 (acc VGPR0, lanes 0..15) = the sums.
// A layout (ISA 7.12.2, 32-bit A 16x4): lanes0-15 M=lane, vgpr0=K0, vgpr1=K1;
// lanes16-31 vgpr0=K2, vgpr1=K3. B 4x16 assumed symmetric half-striping:
// vgpr0 = rows {K0 | K2} across lane halves, vgpr1 = {K1 | K3}.
__global__ __launch_bounds__(128) void stats_wmma_kernel(
        const int*   __restrict__ counts,
        const float* __restrict__ tables,
        const float* __restrict__ gamma,
        const float* __restrict__ beta,
        float* __restrict__ scale,
        float* __restrict__ shift) {
    const int f     = blockIdx.x;
    const int wave  = threadIdx.x >> 5;
    const int lane  = threadIdx.x & 31;
    const int dbase = wave * 16;          // 4 waves cover D=64
    const int n     = lane & 15;          // tile column
    const int hi    = lane >> 4;          // lane half selects K pair

    const float* tf = tables + (size_t)f * V_SZ * D_SZ;
    const int*   cf = counts + f * V_SZ;

    v8f acc1 = {};   // sum c*t
    v8f acc2 = {};   // sum c*t^2

    for (int k = 0; k < V_SZ; k += 4) {
        const int kk = k + 2 * hi;                 // this half's first K row
        const int r0 = min(kk,     V_SZ - 1);      // clamped addresses; tail
        const int r1 = min(kk + 1, V_SZ - 1);      // rows get zero counts

        // A-matrix: only column-0 lanes of each half carry counts (row M=0).
        const float c0 = (float)cf[r0];
        const float c1 = (float)cf[r1];
        v2f a;
        a[0] = (n == 0 && kk     < V_SZ) ? c0 : 0.0f;
        a[1] = (n == 0 && kk + 1 < V_SZ) ? c1 : 0.0f;

        // B-matrix: two table rows, 16 contiguous columns per lane half.
        v2f b, b2;
        b[0] = tf[(size_t)r0 * D_SZ + dbase + n];
        b[1] = tf[(size_t)r1 * D_SZ + dbase + n];
        b2[0] = b[0] * b[0];
        b2[1] = b[1] * b[1];

        acc1 = __builtin_amdgcn_wmma_f32_16x16x4_f32(false, a, false, b,
                                                     (short)0, acc1, false, false);
        acc2 = __builtin_amdgcn_wmma_f32_16x16x4_f32(false, a, false, b2,
                                                     (short)0, acc2, false, false);
    }

    // D row M=0 lives in accumulator VGPR0, lanes 0..15 (N = lane).
    const float S1   = acc1[0];
    const float S2   = acc2[0];
    const float mean = S1 * (1.0f / (float)B_SZ);
    const float var  = S2 * (1.0f / (float)B_SZ) - mean * mean;
    const float rstd = rsqrtf(var + BN_EPS);
    const int   d    = dbase + n;
    const float sc   = gamma[f * D_SZ + d] * rstd;
    const float sh   = beta [f * D_SZ + d] - mean * sc;
    if (lane < 16) {
        scale[f * D_SZ + d] = sc;
        shift[f * D_SZ + d] = sh;
    }
}

// ---------------------------------------------------------------- kernel 4
// Fused gather -> affine BN -> ReLU -> store + numerical tail.
// One block per batch row; thread t covers feature f = t/8, dims
// [8*(t%8), 8*(t%8)+8). Table gathers hit L2 (tables = 8.2 MB << 192 MB);
// output stream (541 MB) uses non-temporal B128 stores to spare L2.
__global__ __launch_bounds__(256) void fuse_kernel(
        const int*   __restrict__ cat_idx,
        const float* __restrict__ numerical,
        const float* __restrict__ tables,
        const float* __restrict__ scale,
        const float* __restrict__ shift,
        float* __restrict__ out) {
    const int b     = blockIdx.x;
    const int t     = threadIdx.x;
    const int f     = t >> 3;
    const int dbase = (t & 7) * 8;

    const int v = cat_idx[b * F_SZ + f];

    const f4* tp  = (const f4*)(tables + ((size_t)f * V_SZ + v) * D_SZ + dbase);
    const f4* scp = (const f4*)(scale + f * D_SZ + dbase);
    const f4* shp = (const f4*)(shift + f * D_SZ + dbase);

    const f4 t0 = tp[0],  t1 = tp[1];
    const f4 s0 = scp[0], s1 = scp[1];
    const f4 h0 = shp[0], h1 = shp[1];

    f4 r0, r1;
#pragma unroll
    for (int j = 0; j < 4; ++j) {
        r0[j] = fmaxf(fmaf(t0[j], s0[j], h0[j]), 0.0f);
        r1[j] = fmaxf(fmaf(t1[j], s1[j], h1[j]), 0.0f);
    }

    float* orow = out + (size_t)b * OUT_COLS;
    __builtin_nontemporal_store(r0, (f4*)(orow + f * D_SZ + dbase));
    __builtin_nontemporal_store(r1, (f4*)(orow + f * D_SZ + dbase + 4));

    if (t < N_NUMC / 4) {
        f4 nv = ((const f4*)(numerical + (size_t)b * N_NUMC))[t];
        __builtin_nontemporal_store(nv, (f4*)(orow + F_SZ * D_SZ + t * 4));
    }
}

// ---------------------------------------------------------------- launch
extern "C" void kernel_launch(void* const* d_in, const int* in_sizes, int n_in,
                              void* d_out, int out_size, void* d_ws, size_t ws_size,
                              hipStream_t stream) {
    const int*   cat_idx   = (const int*)  d_in[0];
    const float* numerical = (const float*)d_in[1];
    const float* tables    = (const float*)d_in[2];
    const float* gamma     = (const float*)d_in[3];
    const float* beta      = (const float*)d_in[4];
    float*       out       = (float*)d_out;

    // Workspace: counts (F*V int32), then scale/shift (F*D f32 each).
    int*   counts = (int*)d_ws;
    size_t coff   = ((size_t)F_SZ * V_SZ * sizeof(int) + 255) & ~(size_t)255;
    float* scale  = (float*)((char*)d_ws + coff);
    float* shift  = scale + F_SZ * D_SZ;

    zero_counts_kernel<<<(F_SZ * V_SZ + 255) / 256, 256, 0, stream>>>(counts);
    hist_kernel<<<(B_SZ * F_SZ + 255) / 256, 256, 0, stream>>>(cat_idx, counts);
    stats_wmma_kernel<<<F_SZ, 128, 0, stream>>>(counts, tables, gamma, beta,
                                                scale, shift);
    fuse_kernel<<<B_SZ, 256, 0, stream>>>(cat_idx, numerical, tables,
                                          scale, shift, out);
}